// Clamp_82592221102246
// MI455X (gfx1250) — compile-verified
//
#include <hip/hip_runtime.h>

// out[i] = x[i] * S[i%6] + T[i%6], fp32, pure streaming (bandwidth bound).
// MI455X: 23.3 TB/s HBM -> ~8.6us floor for 2 x 96MiB. Strategy: 24 floats
// (6 x float4) per thread on 12-float-aligned runs so channel indices are
// compile-time constants; non-temporal B128 loads/stores (single-use data).

typedef float v4f __attribute__((ext_vector_type(4)));

// Per-channel affine constants (fold at compile time in fp32, matching ref).
#define S0 (340.0f / 98.15f)
#define S1 (340.0f / 98.15f)
#define S2 (600.0f / 173.2f)
#define S3 (240.0f / 69.28f)
#define S4 (144.0f / 24.55f)
#define S5 (180.0f / 51.96f)

#define T0 (-170.0f / 98.15f)
#define T1 (-170.0f / 98.15f)
#define T2 (-300.0f / 173.2f)
#define T3 ((60.0f - 180.0f) / 69.28f)
#define T4 ((6.0f - 36.66f) / 24.55f)
#define T5 (-90.0f / 51.96f)

__device__ __forceinline__ v4f affine4(v4f a, float s0, float t0, float s1, float t1,
                                       float s2, float t2, float s3, float t3) {
    v4f r;
    r.x = __builtin_fmaf(a.x, s0, t0);
    r.y = __builtin_fmaf(a.y, s1, t1);
    r.z = __builtin_fmaf(a.z, s2, t2);
    r.w = __builtin_fmaf(a.w, s3, t3);
    return r;
}

// Apply the fixed 12-element channel pattern (0..5,0..5) to three float4s.
__device__ __forceinline__ void affine12(const v4f a[3], v4f r[3]) {
    r[0] = affine4(a[0], S0, T0, S1, T1, S2, T2, S3, T3);
    r[1] = affine4(a[1], S4, T4, S5, T5, S0, T0, S1, T1);
    r[2] = affine4(a[2], S2, T2, S3, T3, S4, T4, S5, T5);
}

__global__ __launch_bounds__(256) void affine6_stream_kernel(
    const float* __restrict__ x, float* __restrict__ y, long long n) {
    const long long t = (long long)blockIdx.x * blockDim.x + threadIdx.x;
    const long long base = t * 24;  // 24 floats (= two 12-channel groups) per thread

    if (base + 24 <= n) {
        const v4f* __restrict__ xi = (const v4f*)(x + base);
        v4f* __restrict__ yo = (v4f*)(y + base);

        v4f a[6];
#pragma unroll
        for (int i = 0; i < 6; ++i)
            a[i] = __builtin_nontemporal_load(xi + i);  // global_load_b128 th:NT

        v4f r[6];
        affine12(&a[0], &r[0]);
        affine12(&a[3], &r[3]);

#pragma unroll
        for (int i = 0; i < 6; ++i)
            __builtin_nontemporal_store(r[i], yo + i);  // global_store_b128 th:NT
    } else if (base < n) {
        // Scalar tail (at most one thread takes this path).
        const float S[6] = {S0, S1, S2, S3, S4, S5};
        const float T[6] = {T0, T1, T2, T3, T4, T5};
        for (long long i = base; i < n; ++i) {
            int k = (int)(i % 6);
            y[i] = __builtin_fmaf(x[i], S[k], T[k]);
        }
    }
}

extern "C" void kernel_launch(void* const* d_in, const int* in_sizes, int n_in,
                              void* d_out, int out_size, void* d_ws, size_t ws_size,
                              hipStream_t stream) {
    (void)n_in; (void)d_ws; (void)ws_size;
    const float* x = (const float*)d_in[0];
    float* y = (float*)d_out;
    const long long n = (long long)in_sizes[0];  // == out_size

    const long long threads_needed = (n + 23) / 24;
    const int block = 256;
    const long long grid = (threads_needed + block - 1) / block;

    affine6_stream_kernel<<<(unsigned)grid, block, 0, stream>>>(x, y, n);
}